// PairwiseRefine_28819230556872
// MI455X (gfx1250) — compile-verified
//
#include <hip/hip_runtime.h>

// Problem constants (from reference): B=16, K=256, D=256, H=128
#define BQ 16
#define KQ 256
#define DQ 256
#define HQ 128
#define MQ (BQ * KQ) // 4096 flattened rows

typedef float v2f __attribute__((ext_vector_type(2)));
typedef float v8f __attribute__((ext_vector_type(8)));
typedef int   v4i __attribute__((ext_vector_type(4)));

#if __has_builtin(__builtin_amdgcn_global_load_async_to_lds_b128) && \
    __has_builtin(__builtin_amdgcn_s_wait_asynccnt)
#define USE_ASYNC_LDS 1
typedef __attribute__((address_space(1))) v4i* gv4i_p; // global int4*
typedef __attribute__((address_space(3))) v4i* lv4i_p; // LDS int4*
#else
#define USE_ASYNC_LDS 0
#endif

// ---------------------------------------------------------------------------
// Kernel 1: Aout = x @ W1[:256], Bout = x @ W1[256:]
// One wave computes one 16x16 output tile via v_wmma_f32_16x16x4_f32,
// looping K in steps of 4. grid.y selects which half of W1 / which output.
// ---------------------------------------------------------------------------
__global__ __launch_bounds__(256) void k_proj(const float* __restrict__ x,
                                              const float* __restrict__ W1,
                                              float* __restrict__ Aout,
                                              float* __restrict__ Bout) {
  const int wave = threadIdx.x >> 5;
  const int lane = threadIdx.x & 31;
  const int half = lane >> 4;   // which K-pair this lane holds
  const int r    = lane & 15;   // row (A) / col (B,C)

  const int tile = blockIdx.x * 8 + wave;     // 2048 tiles total
  const int tn = tile & 7;                    // 128/16 = 8 N-tiles
  const int tm = tile >> 3;                   // 256 M-tiles
  const int m0 = tm * 16, n0 = tn * 16;

  const float* Wp = W1 + (size_t)blockIdx.y * DQ * HQ;
  float* Cp = (blockIdx.y == 0) ? Aout : Bout;

  v8f c = {};
  const float* arow = x  + (size_t)(m0 + r) * DQ + 2 * half;
  const float* bcol = Wp + (size_t)(2 * half) * HQ + n0 + r;
  for (int kk = 0; kk < DQ; kk += 4) {
    v2f a; a.x = arow[kk];                 a.y = arow[kk + 1];
    v2f b; b.x = bcol[(size_t)kk * HQ];    b.y = bcol[(size_t)(kk + 1) * HQ];
    c = __builtin_amdgcn_wmma_f32_16x16x4_f32(false, a, false, b,
                                              (short)0, c, false, false);
  }

  float* crow = Cp + (size_t)(m0 + 8 * half) * HQ + n0 + r;
#pragma unroll
  for (int i = 0; i < 8; ++i) crow[(size_t)i * HQ] = c[i];
}

// ---------------------------------------------------------------------------
// Kernel 2: mh[b,k,h] = (1/K) * sum_j relu(A[b,k,h] + Bv[b,j,h] + b1[h])
// One block per (batch, 16-row k-tile). Bv slab streamed through two 32KB
// LDS buffers (64 j-rows each) with async global->LDS double buffering:
// chunk p+1 is in flight on the ASYNC pipe while the VALU loop runs chunk p.
// Each thread keeps 8 accumulators so one LDS read of b_j feeds 8 outputs.
// ---------------------------------------------------------------------------
#define CHUNK_ROWS 64
#define CHUNK_F4   ((CHUNK_ROWS * HQ) / 4)   // 2048 float4 per chunk

// Issue one 32KB chunk copy: 8 x b128 per thread (async if available).
__device__ __forceinline__ void issue_chunk(const float* __restrict__ gsrc,
                                            float* __restrict__ lbuf, int tid) {
  const float4* g = (const float4*)gsrc;
  float4* l = (float4*)lbuf;
#if USE_ASYNC_LDS
#pragma unroll
  for (int it = 0; it < 8; ++it) {
    __builtin_amdgcn_global_load_async_to_lds_b128(
        (gv4i_p)(g + tid + it * 256), (lv4i_p)(l + tid + it * 256), 0, 0);
  }
#else
#pragma unroll
  for (int it = 0; it < 8; ++it) l[tid + it * 256] = g[tid + it * 256];
#endif
}

__global__ __launch_bounds__(256) void k_pair(const float* __restrict__ Aout,
                                              const float* __restrict__ Bv,
                                              const float* __restrict__ b1,
                                              float* __restrict__ mh) {
  __shared__ float Bsh[2][CHUNK_ROWS * HQ]; // 2 x 32 KB
  const int tid = threadIdx.x;
  const int bb = blockIdx.x >> 4;          // batch
  const int k0 = (blockIdx.x & 15) * 16;   // k tile base
  const int h = tid & (HQ - 1);
  const int kbase = tid >> 7;              // 0 or 1

  const float* base = Bv + (size_t)(bb * KQ) * HQ;
  issue_chunk(base, Bsh[0], tid);          // chunk 0 in flight

  float avals[8], accs[8];
#pragma unroll
  for (int i = 0; i < 8; ++i) {
    const int k = k0 + kbase + 2 * i;
    avals[i] = Aout[(size_t)(bb * KQ + k) * HQ + h] + b1[h];
    accs[i] = 0.0f;
  }

  for (int p = 0; p < 4; ++p) {
    if (p < 3)
      issue_chunk(base + (size_t)(p + 1) * CHUNK_ROWS * HQ, Bsh[(p + 1) & 1], tid);
#if USE_ASYNC_LDS
    // In-order completion: waiting until only the next chunk's 8 ops remain
    // guarantees chunk p has fully landed in LDS.
    if (p < 3) __builtin_amdgcn_s_wait_asynccnt(8);
    else       __builtin_amdgcn_s_wait_asynccnt(0);
#endif
    __syncthreads();

    const float* buf = Bsh[p & 1];
#pragma unroll 4
    for (int j = 0; j < CHUNK_ROWS; ++j) {
      const float bj = buf[j * HQ + h];
#pragma unroll
      for (int i = 0; i < 8; ++i) accs[i] += fmaxf(avals[i] + bj, 0.0f);
    }
    __syncthreads();
  }

  const float inv = 1.0f / (float)KQ;
#pragma unroll
  for (int i = 0; i < 8; ++i) {
    const int k = k0 + kbase + 2 * i;
    mh[(size_t)(bb * KQ + k) * HQ + h] = accs[i] * inv;
  }
}

// ---------------------------------------------------------------------------
// Kernel 3: out = mh @ W2 + b2 + x  (WMMA f32, K=128 -> 32 steps/tile)
// ---------------------------------------------------------------------------
__global__ __launch_bounds__(256) void k_out(const float* __restrict__ mh,
                                             const float* __restrict__ W2,
                                             const float* __restrict__ b2,
                                             const float* __restrict__ x,
                                             float* __restrict__ out) {
  const int wave = threadIdx.x >> 5;
  const int lane = threadIdx.x & 31;
  const int half = lane >> 4;
  const int r    = lane & 15;

  const int tile = blockIdx.x * 8 + wave;     // 4096 tiles
  const int tn = tile & 15;                   // 256/16 = 16 N-tiles
  const int tm = tile >> 4;                   // 256 M-tiles
  const int m0 = tm * 16, n0 = tn * 16;

  v8f c = {};
  const float* arow = mh + (size_t)(m0 + r) * HQ + 2 * half;
  const float* bcol = W2 + (size_t)(2 * half) * DQ + n0 + r;
  for (int kk = 0; kk < HQ; kk += 4) {
    v2f a; a.x = arow[kk];                 a.y = arow[kk + 1];
    v2f b; b.x = bcol[(size_t)kk * DQ];    b.y = bcol[(size_t)(kk + 1) * DQ];
    c = __builtin_amdgcn_wmma_f32_16x16x4_f32(false, a, false, b,
                                              (short)0, c, false, false);
  }

  const float bias = b2[n0 + r];
  const float* xrow = x   + (size_t)(m0 + 8 * half) * DQ + n0 + r;
  float*       crow = out + (size_t)(m0 + 8 * half) * DQ + n0 + r;
#pragma unroll
  for (int i = 0; i < 8; ++i)
    crow[(size_t)i * DQ] = c[i] + bias + xrow[(size_t)i * DQ];
}

// ---------------------------------------------------------------------------
extern "C" void kernel_launch(void* const* d_in, const int* in_sizes, int n_in,
                              void* d_out, int out_size, void* d_ws, size_t ws_size,
                              hipStream_t stream) {
  const float* x  = (const float*)d_in[0];
  const float* W1 = (const float*)d_in[1];
  const float* b1 = (const float*)d_in[2];
  const float* W2 = (const float*)d_in[3];
  const float* b2 = (const float*)d_in[4];
  float* out = (float*)d_out;

  float* Aout = (float*)d_ws;                    // 4096*128 f32 = 2 MB
  float* Bout = Aout + (size_t)MQ * HQ;          // 2 MB
  float* mh   = Bout + (size_t)MQ * HQ;          // 2 MB

  // 2048 tiles / 8 waves per block = 256 blocks, y selects W1a/W1b
  k_proj<<<dim3(256, 2), 256, 0, stream>>>(x, W1, Aout, Bout);
  // one block per (batch, 16-k tile) = 256 blocks
  k_pair<<<dim3(256), 256, 0, stream>>>(Aout, Bout, b1, mh);
  // 4096 tiles / 8 waves per block = 512 blocks
  k_out<<<dim3(512), 256, 0, stream>>>(mh, W2, b2, x, out);
}